// SparseGraphAttention_65755949301762
// MI455X (gfx1250) — compile-verified
//
#include <hip/hip_runtime.h>
#include <hip/hip_bf16.h>

#define GAT_H 8
#define GAT_FIN 128
#define GAT_FOUT 64
#define GAT_ALPHA 0.2f

typedef __attribute__((ext_vector_type(2))) float v2f;
typedef __attribute__((ext_vector_type(8))) float v8f;

// ---------------------------------------------------------------------------
// K1: zero the atomic accumulators (d_out and rowsum), float4-vectorized.
// d_out / d_ws are poisoned by the harness -> must clear before scatter-add.
// Covers n*512 floats of out (n*128 float4) and n*8 floats of rowsum (n*2 f4).
// ---------------------------------------------------------------------------
__global__ void gat_zero_kernel(float4* __restrict__ out, float4* __restrict__ rowsum, int n) {
  int idx = blockIdx.x * blockDim.x + threadIdx.x;
  const float4 z = make_float4(0.f, 0.f, 0.f, 0.f);
  if (idx < n * (GAT_H * GAT_FOUT / 4)) out[idx] = z;
  if (idx < n * (GAT_H / 4)) rowsum[idx] = z;
}

// ---------------------------------------------------------------------------
// K2: h[n][h][o] = x[n][:] @ W[h][:][o], fp32 WMMA (V_WMMA_F32_16X16X4_F32).
// grid = (N/16 row tiles, H), block = 128 (4 waves); wave w -> col tile w*16.
// A 16x4 f32 layout: lane(0..15)=M, vgpr0/1 = K+0/K+1, lanes 16..31 = K+2/K+3.
// B 4x16 f32 layout: lane(0..15)=N, vgpr0/1 = K+0/K+1, lanes 16..31 = K+2/K+3.
// C/D: vgpr j holds M = j (lanes 0..15) / j+8 (lanes 16..31), N = lane%16.
// N = 50000 = 3125 * 16 exactly -> no bounds checks, EXEC stays all-ones.
// ---------------------------------------------------------------------------
__global__ void gat_gemm_kernel(const float* __restrict__ x,
                                const float* __restrict__ W,
                                float* __restrict__ hbuf) {
  const int lane = threadIdx.x & 31;
  const int wave = threadIdx.x >> 5;          // 0..3 -> column tile
  const int row0 = blockIdx.x * 16;
  const int head = blockIdx.y;
  const int col0 = wave * 16;
  const int m    = lane & 15;                 // M (A) or N (B) within tile
  const int koff = (lane >> 4) * 2;           // upper half-wave holds K+2,K+3

  const float* xrow = x + (size_t)(row0 + m) * GAT_FIN + koff;
  const float* wcol = W + (size_t)head * GAT_FIN * GAT_FOUT
                        + (size_t)koff * GAT_FOUT + col0 + m;

  v8f c = {};
#pragma unroll
  for (int k = 0; k < GAT_FIN; k += 4) {
    v2f afrag = *(const v2f*)(xrow + k);      // x[row][k+koff], x[row][k+koff+1]
    v2f bfrag;
    bfrag.x = wcol[(k + 0) * GAT_FOUT];       // W[head][k+koff  ][col]
    bfrag.y = wcol[(k + 1) * GAT_FOUT];       // W[head][k+koff+1][col]
    c = __builtin_amdgcn_wmma_f32_16x16x4_f32(
        /*neg_a=*/false, afrag, /*neg_b=*/false, bfrag,
        /*c_mod=*/(short)0, c, /*reuse_a=*/false, /*reuse_b=*/false);
  }

  // Store: hbuf layout [n][head][o], row stride = H*FOUT = 512 floats.
  float* outp = hbuf + ((size_t)(row0 + 8 * (lane >> 4)) * (GAT_H * GAT_FOUT))
                     + (size_t)head * GAT_FOUT + col0 + m;
#pragma unroll
  for (int j = 0; j < 8; j++) {
    outp[(size_t)j * (GAT_H * GAT_FOUT)] = c[j];
  }
}

// ---------------------------------------------------------------------------
// K3: per (n, head) attention scores: ssrc = h . a[:64], sdst = h . a[64:].
// float4-vectorized loads (b128).
// ---------------------------------------------------------------------------
__global__ void gat_score_kernel(const float* __restrict__ hbuf,
                                 const float* __restrict__ a,
                                 float* __restrict__ ssrc,
                                 float* __restrict__ sdst, int n) {
  int idx = blockIdx.x * blockDim.x + threadIdx.x;   // over N*H
  if (idx >= n * GAT_H) return;
  int node = idx >> 3;
  int head = idx & 7;
  const float4* hp = (const float4*)(hbuf + (size_t)node * (GAT_H * GAT_FOUT) + head * GAT_FOUT);
  const float4* as = (const float4*)(a + head * (2 * GAT_FOUT));
  const float4* ad = as + (GAT_FOUT / 4);
  float s0 = 0.0f, s1 = 0.0f;
#pragma unroll
  for (int o = 0; o < GAT_FOUT / 4; o++) {
    float4 v = hp[o];
    float4 u = as[o];
    float4 w = ad[o];
    s0 += v.x * u.x + v.y * u.y + v.z * u.z + v.w * u.w;
    s1 += v.x * w.x + v.y * w.y + v.z * w.z + v.w * w.w;
  }
  ssrc[idx] = s0;
  sdst[idx] = s1;
}

// ---------------------------------------------------------------------------
// K4: per-edge softmax weight + scatter-add of weighted messages.
// One 128-thread block per edge. src/dst are block-uniform -> scalar loads.
// Each thread gathers one float4 of h[dst] (global_load_b128) and issues
// 4 global_atomic_add_f32 into out[src]. 8 lanes compute per-head weights.
// ---------------------------------------------------------------------------
__global__ void gat_edge_kernel(const int* __restrict__ src,
                                const int* __restrict__ dst,
                                const float* __restrict__ hbuf,
                                const float* __restrict__ ssrc,
                                const float* __restrict__ sdst,
                                float* __restrict__ rowsum,
                                float* __restrict__ out) {
  const int e = blockIdx.x;
  const int t = threadIdx.x;                  // 0..127
  const int sn = src[e];
  const int dn = dst[e];

  __shared__ float wsh[GAT_H];
  if (t < GAT_H) {
    float s  = ssrc[sn * GAT_H + t] + sdst[dn * GAT_H + t];
    float ls = (s > 0.0f) ? s : (GAT_ALPHA * s);
    float we = __expf(-ls);
    wsh[t] = we;
    atomicAdd(&rowsum[sn * GAT_H + t], we);
  }
  __syncthreads();

  // 512 floats per edge = 128 float4s; head = t/16 (64 floats = 16 float4 per head)
  const float4 hv = ((const float4*)(hbuf + (size_t)dn * (GAT_H * GAT_FOUT)))[t];
  const float  w  = wsh[t >> 4];
  float* op = out + (size_t)sn * (GAT_H * GAT_FOUT) + t * 4;
  atomicAdd(op + 0, w * hv.x);
  atomicAdd(op + 1, w * hv.y);
  atomicAdd(op + 2, w * hv.z);
  atomicAdd(op + 3, w * hv.w);
}

// ---------------------------------------------------------------------------
// K5: out /= rowsum (in place), float4-vectorized.
// ---------------------------------------------------------------------------
__global__ void gat_norm_kernel(float4* __restrict__ out,
                                const float* __restrict__ rowsum, int n) {
  int idx = blockIdx.x * blockDim.x + threadIdx.x;   // over N*128 float4s
  if (idx >= n * (GAT_H * GAT_FOUT / 4)) return;
  int node = idx >> 7;                                // /128
  int head = (idx >> 4) & 7;                          // /16 % 8
  float rs = rowsum[node * GAT_H + head];
  float4 v = out[idx];
  v.x /= rs; v.y /= rs; v.z /= rs; v.w /= rs;
  out[idx] = v;
}

// ---------------------------------------------------------------------------
extern "C" void kernel_launch(void* const* d_in, const int* in_sizes, int n_in,
                              void* d_out, int out_size, void* d_ws, size_t ws_size,
                              hipStream_t stream) {
  const float* x    = (const float*)d_in[0];
  const int*   edge = (const int*)d_in[1];
  const float* W    = (const float*)d_in[2];
  const float* a    = (const float*)d_in[3];
  float*       out  = (float*)d_out;

  const int N = in_sizes[0] / GAT_FIN;   // 50000
  const int E = in_sizes[1] / 2;         // 800000
  const int* src = edge;                 // edge row 0
  const int* dst = edge + E;             // edge row 1

  // Workspace carve-up (floats): h [N*512] | ssrc [N*8] | sdst [N*8] | rowsum [N*8]
  float* hbuf   = (float*)d_ws;
  float* ssrc   = hbuf + (size_t)N * (GAT_H * GAT_FOUT);
  float* sdst   = ssrc + (size_t)N * GAT_H;
  float* rowsum = sdst + (size_t)N * GAT_H;

  const int out_v4 = N * (GAT_H * GAT_FOUT / 4);     // 6.4M float4

  gat_zero_kernel<<<(out_v4 + 255) / 256, 256, 0, stream>>>(
      (float4*)out, (float4*)rowsum, N);

  dim3 ggrid(N / 16, GAT_H);             // 50000 % 16 == 0
  gat_gemm_kernel<<<ggrid, 128, 0, stream>>>(x, W, hbuf);

  gat_score_kernel<<<(N * GAT_H + 255) / 256, 256, 0, stream>>>(hbuf, a, ssrc, sdst, N);

  gat_edge_kernel<<<E, 128, 0, stream>>>(src, dst, hbuf, ssrc, sdst, rowsum, out);

  gat_norm_kernel<<<(out_v4 + 255) / 256, 256, 0, stream>>>((float4*)out, rowsum, N);
}